// GIN_72971494359663
// MI455X (gfx1250) — compile-verified
//
#include <hip/hip_runtime.h>
#include <math.h>

#define N_NODES_C   50000
#define N_EDGES_C   800000
#define NUM_GRAPHS_C 512
#define D_H_C       64
#define BN_EPS_C    1e-5f

typedef __attribute__((ext_vector_type(2))) float v2f;
typedef __attribute__((ext_vector_type(8))) float v8f;

// ---------------- workspace layout (floats) ----------------
// [0, 3.2M)          agg
// [3.2M, 6.4M)       h1 (pre-BN hidden)
// [6.4M, +64)        colsum
// [+64, +128)        colsumsq
// [+128, +192)       bn scale
// [+192, +256)       bn shift
// [6.4M+256, +65536) pool g[512][128]  (cols 0..63 add-pool, 64..127 max-pool bits)
#define OFF_AGG   0
#define OFF_H1    3200000
#define OFF_STAT  6400000
#define OFF_POOL  6400256

// ---------------------------------------------------------------------------
__global__ __launch_bounds__(256) void k_init_agg(const float* __restrict__ x,
                                                  float* __restrict__ agg) {
    int t = blockIdx.x * 256 + threadIdx.x;   // 800000 threads, float4 each
    ((float4*)agg)[t] = ((const float4*)x)[t];
}

__global__ __launch_bounds__(256) void k_zero_aux(float* __restrict__ p, int n) {
    int t = blockIdx.x * 256 + threadIdx.x;
    if (t < n) p[t] = 0.0f;
}

// one edge handled by 16 lanes (4 floats each)
__global__ __launch_bounds__(256) void k_edge_agg(const float* __restrict__ x,
                                                  const int* __restrict__ src,
                                                  const int* __restrict__ dst,
                                                  float* __restrict__ agg) {
    int t  = blockIdx.x * 256 + threadIdx.x;  // 50000 blocks -> 800000*16 units
    int e  = t >> 4;
    int c4 = (t & 15) << 2;
    int s = src[e], d = dst[e];
    float4 v = *(const float4*)(x + (size_t)s * 64 + c4);
    float* p = agg + (size_t)d * 64 + c4;
    atomicAdd(p + 0, v.x);
    atomicAdd(p + 1, v.y);
    atomicAdd(p + 2, v.z);
    atomicAdd(p + 3, v.w);
}

// ---------------------------------------------------------------------------
// GEMM1: h1 = agg @ W1 + b1, fused per-column sum / sumsq partials (BN stats)
// block = 128 threads = 4 waves; wave w owns output cols [16w,16w+16); 16 rows/block
__global__ __launch_bounds__(128) void k_gemm1_stats(const float* __restrict__ agg,
                                                     const float* __restrict__ W1,
                                                     const float* __restrict__ b1,
                                                     float* __restrict__ h1,
                                                     float* __restrict__ colsum,
                                                     float* __restrict__ colsumsq) {
    __shared__ float Ws[64 * 64];
    __shared__ float bs[64];
    for (int i = threadIdx.x; i < 64 * 64; i += 128) Ws[i] = W1[i];
    if (threadIdx.x < 64) bs[threadIdx.x] = b1[threadIdx.x];
    __syncthreads();

    const int wave = threadIdx.x >> 5;
    const int lane = threadIdx.x & 31;
    const int n0   = wave * 16;
    const int m0   = blockIdx.x * 16;
    const int m    = lane & 15;       // A row (in tile) / B col (in tile)
    const int half = lane >> 4;       // K sub-pair select
    const int col  = n0 + m;

    v8f c;
    {
        float bias = bs[col];
        #pragma unroll
        for (int r = 0; r < 8; ++r) c[r] = bias;
    }
    const float* arow = agg + (size_t)(m0 + m) * 64;
    #pragma unroll
    for (int kk = 0; kk < 64; kk += 4) {
        int k = kk + 2 * half;
        v2f a, b;
        a.x = arow[k];
        a.y = arow[k + 1];
        b.x = Ws[k * 64 + col];
        b.y = Ws[(k + 1) * 64 + col];
        c = __builtin_amdgcn_wmma_f32_16x16x4_f32(false, a, false, b,
                                                  (short)0, c, false, false);
    }
    float s = 0.f, sq = 0.f;
    #pragma unroll
    for (int r = 0; r < 8; ++r) {
        int row = m0 + r + 8 * half;
        float v = c[r];
        h1[(size_t)row * 64 + col] = v;
        s  += v;
        sq += v * v;
    }
    // lanes L and L+16 hold the same column -> fold, then one atomic per col
    s  += __shfl_xor(s, 16, 32);
    sq += __shfl_xor(sq, 16, 32);
    if (half == 0) {
        atomicAdd(&colsum[col], s);
        atomicAdd(&colsumsq[col], sq);
    }
}

__global__ void k_bn_finalize(const float* __restrict__ colsum,
                              const float* __restrict__ colsumsq,
                              const float* __restrict__ gamma,
                              const float* __restrict__ beta,
                              float* __restrict__ scale,
                              float* __restrict__ shift) {
    int c = threadIdx.x;
    if (c < 64) {
        const float invN = 1.0f / (float)N_NODES_C;
        float mean = colsum[c] * invN;
        float var  = colsumsq[c] * invN - mean * mean;
        float inv  = rsqrtf(var + BN_EPS_C);
        float sc   = gamma[c] * inv;
        scale[c] = sc;
        shift[c] = beta[c] - mean * sc;
    }
}

// ---------------------------------------------------------------------------
// GEMM2: h2 = relu( relu(BN(h1)) @ W2 + b2 ), fused add/max pooling per graph.
__global__ __launch_bounds__(128) void k_gemm2_pool(const float* __restrict__ h1,
                                                    const float* __restrict__ scale,
                                                    const float* __restrict__ shift,
                                                    const float* __restrict__ W2,
                                                    const float* __restrict__ b2,
                                                    const int* __restrict__ batch,
                                                    float* __restrict__ pool) {
    __shared__ float Ws[64 * 64];
    __shared__ float bs[64], scs[64], shs[64];
    for (int i = threadIdx.x; i < 64 * 64; i += 128) Ws[i] = W2[i];
    if (threadIdx.x < 64) {
        bs[threadIdx.x]  = b2[threadIdx.x];
        scs[threadIdx.x] = scale[threadIdx.x];
        shs[threadIdx.x] = shift[threadIdx.x];
    }
    __syncthreads();

    const int wave = threadIdx.x >> 5;
    const int lane = threadIdx.x & 31;
    const int n0   = wave * 16;
    const int m0   = blockIdx.x * 16;
    const int m    = lane & 15;
    const int half = lane >> 4;
    const int col  = n0 + m;

    v8f c;
    {
        float bias = bs[col];
        #pragma unroll
        for (int r = 0; r < 8; ++r) c[r] = bias;
    }
    const float* arow = h1 + (size_t)(m0 + m) * 64;
    #pragma unroll
    for (int kk = 0; kk < 64; kk += 4) {
        int k = kk + 2 * half;
        v2f a, b;
        a.x = fmaxf(arow[k]     * scs[k]     + shs[k],     0.0f);   // BN + ReLU
        a.y = fmaxf(arow[k + 1] * scs[k + 1] + shs[k + 1], 0.0f);
        b.x = Ws[k * 64 + col];
        b.y = Ws[(k + 1) * 64 + col];
        c = __builtin_amdgcn_wmma_f32_16x16x4_f32(false, a, false, b,
                                                  (short)0, c, false, false);
    }
    #pragma unroll
    for (int r = 0; r < 8; ++r) {
        int   row = m0 + r + 8 * half;
        float v   = fmaxf(c[r], 0.0f);
        int   g   = batch[row];
        float* gp = pool + (size_t)g * 128;
        atomicAdd(gp + col, v);
        // max-pool: v >= 0 so IEEE bits order as uints; pool init = 0.0f
        atomicMax((unsigned int*)(gp + 64 + col), __float_as_uint(v));
    }
}

// ---------------------------------------------------------------------------
// Final MLP: gg = relu(g @ Wl1 + bl1); logits = gg @ Wl2 + bl2; out = sigmoid
// block = 256 threads = 8 waves (one n-tile each), 16 graphs per block.
__global__ __launch_bounds__(256) void k_final_mlp(const float* __restrict__ pool,
                                                   const float* __restrict__ Wl1,
                                                   const float* __restrict__ bl1,
                                                   const float* __restrict__ Wl2,
                                                   const float* __restrict__ bl2,
                                                   float* __restrict__ out) {
    __shared__ float gg[16 * 128];
    const int wave = threadIdx.x >> 5;
    const int lane = threadIdx.x & 31;
    const int n0   = wave * 16;
    const int m0   = blockIdx.x * 16;
    const int m    = lane & 15;
    const int half = lane >> 4;
    const int col  = n0 + m;

    v8f c;
    {
        float bias = bl1[col];
        #pragma unroll
        for (int r = 0; r < 8; ++r) c[r] = bias;
    }
    const float* grow = pool + (size_t)(m0 + m) * 128;
    #pragma unroll
    for (int kk = 0; kk < 128; kk += 4) {
        int k = kk + 2 * half;
        v2f a, b;
        a.x = grow[k];
        a.y = grow[k + 1];
        b.x = Wl1[k * 128 + col];
        b.y = Wl1[(k + 1) * 128 + col];
        c = __builtin_amdgcn_wmma_f32_16x16x4_f32(false, a, false, b,
                                                  (short)0, c, false, false);
    }
    #pragma unroll
    for (int r = 0; r < 8; ++r) {
        int row = r + 8 * half;
        gg[row * 128 + col] = fmaxf(c[r], 0.0f);
    }
    __syncthreads();

    if (threadIdx.x < 16) {
        float acc = bl2[0];
        #pragma unroll 4
        for (int j = 0; j < 128; ++j) acc += gg[threadIdx.x * 128 + j] * Wl2[j];
        out[512 + m0 + threadIdx.x] = acc;                        // logits
        out[m0 + threadIdx.x]       = 1.0f / (1.0f + expf(-acc)); // sigmoid
    }
}

// ---------------------------------------------------------------------------
extern "C" void kernel_launch(void* const* d_in, const int* in_sizes, int n_in,
                              void* d_out, int out_size, void* d_ws, size_t ws_size,
                              hipStream_t stream) {
    const float* x     = (const float*)d_in[0];
    const int*   eidx  = (const int*)d_in[1];     // [2, 800000]
    const int*   batch = (const int*)d_in[2];
    const float* W1    = (const float*)d_in[3];
    const float* b1    = (const float*)d_in[4];
    const float* gamma = (const float*)d_in[5];
    const float* beta  = (const float*)d_in[6];
    const float* W2    = (const float*)d_in[7];
    const float* b2    = (const float*)d_in[8];
    const float* Wl1   = (const float*)d_in[9];
    const float* bl1   = (const float*)d_in[10];
    const float* Wl2   = (const float*)d_in[11];
    const float* bl2   = (const float*)d_in[12];
    float*       out   = (float*)d_out;

    float* ws       = (float*)d_ws;
    float* agg      = ws + OFF_AGG;
    float* h1       = ws + OFF_H1;
    float* colsum   = ws + OFF_STAT;
    float* colsumsq = ws + OFF_STAT + 64;
    float* bnscale  = ws + OFF_STAT + 128;
    float* bnshift  = ws + OFF_STAT + 192;
    float* pool     = ws + OFF_POOL;

    const int* src = eidx;
    const int* dst = eidx + N_EDGES_C;

    // 1) agg = x ; zero stats + pool
    k_init_agg<<<N_EDGES_C / 256 * 256 ? (N_NODES_C * 64 / 4 + 255) / 256 : 0,
                 256, 0, stream>>>(x, agg);
    {
        int nz = 256 + NUM_GRAPHS_C * 128;   // stats block + pool
        k_zero_aux<<<(nz + 255) / 256, 256, 0, stream>>>(colsum, nz);
    }
    // 2) scatter-add over edges
    k_edge_agg<<<(N_EDGES_C * 16) / 256, 256, 0, stream>>>(x, src, dst, agg);
    // 3) GEMM1 + BN stats
    k_gemm1_stats<<<N_NODES_C / 16, 128, 0, stream>>>(agg, W1, b1, h1,
                                                      colsum, colsumsq);
    // 4) BN finalize
    k_bn_finalize<<<1, 64, 0, stream>>>(colsum, colsumsq, gamma, beta,
                                        bnscale, bnshift);
    // 5) BN+ReLU -> GEMM2 -> ReLU -> pooling
    k_gemm2_pool<<<N_NODES_C / 16, 128, 0, stream>>>(h1, bnscale, bnshift,
                                                     W2, b2, batch, pool);
    // 6) final MLP + sigmoid
    k_final_mlp<<<NUM_GRAPHS_C / 16, 256, 0, stream>>>(pool, Wl1, bl1,
                                                       Wl2, bl2, out);
}